// RescorlaWagnerModel_7670811590768
// MI455X (gfx1250) — compile-verified
//
#include <hip/hip_runtime.h>
#include <stdint.h>

// Rescorla-Wagner scan: N=8192 tasks, T=1000 trials, K=3 options.
// One thread per task; recurrent state in registers. Bandwidth floor ~8.5us
// (197 MB @ 23.3 TB/s); only 256 waves exist device-wide, so input streams are
// staged into LDS one full stage ahead with gfx1250 async-to-LDS DMA
// (global_load_async_to_lds_b128 + s_wait_asynccnt double-buffering) to make
// latency hiding deterministic. Single-wave workgroups (32 threads) x 256
// blocks spread the 256 waves across the maximum number of WGPs, since the
// kernel is VALU-issue-bound per wave. Output: 128-bit nontemporal stores.

#define TT      1000
#define KK      3
#define BLK     32                 // threads per block (1 wave -> 256 blocks)
#define STRIALS 40                 // trials per stage = 160B per stream per thread
#define NSTAGE  (TT / STRIALS)     // 25 stages, exact
#define SCHUNK  (STRIALS / 4)      // 10 x 128-bit chunks per stage
#define AOPS    30                 // async ops issued per stage (3 * SCHUNK)

typedef float v4f __attribute__((ext_vector_type(4)));
typedef int   v4i __attribute__((ext_vector_type(4)));

#define HAVE_ASYNC 0
#define HAVE_WAITA 0
#if defined(__has_builtin)
#if __has_builtin(__builtin_amdgcn_global_load_async_to_lds_b128)
#undef HAVE_ASYNC
#define HAVE_ASYNC 1
#endif
#if __has_builtin(__builtin_amdgcn_s_wait_asynccnt)
#undef HAVE_WAITA
#define HAVE_WAITA 1
#endif
#endif

// builtin signature (from clang diagnostics): (v4i AS1* gsrc, v4i AS3* ldst,
// imm offset, imm cpol). AS1 via 64-bit int cast; AS3 = low 32 bits of the
// generic shared-memory address (flat LDS aperture keeps offset in [31:0]).
typedef __attribute__((address_space(1))) v4i as1_v4i;
typedef __attribute__((address_space(3))) v4i as3_v4i;

#if HAVE_ASYNC
#define ASYNC_COPY16(g, l)                                        \
    __builtin_amdgcn_global_load_async_to_lds_b128(               \
        (as1_v4i*)(uintptr_t)(g),                                 \
        (as3_v4i*)(unsigned)(uintptr_t)(l), 0, 0)
#else
#define ASYNC_COPY16(g, l) (*(v4i*)(l) = *(const v4i*)(g))
#endif

#if HAVE_ASYNC && HAVE_WAITA
#define WAIT_ASYNC_LE(imm) __builtin_amdgcn_s_wait_asynccnt(imm)
#elif HAVE_ASYNC
#define WAIT_ASYNC_LE(imm) asm volatile("s_wait_asynccnt %0" :: "i"(imm))
#else
#define WAIT_ASYNC_LE(imm) ((void)0)
#endif

__global__ __launch_bounds__(BLK, 1) void rw_scan_kernel(
    const int*   __restrict__ choice,
    const int*   __restrict__ choice_upd,
    const float* __restrict__ reward,
    const float* __restrict__ p_ap,
    const float* __restrict__ p_am,
    const float* __restrict__ p_iv,
    const float* __restrict__ p_bi,
    const float* __restrict__ p_bs,
    const float* __restrict__ p_bt,
    float*       __restrict__ out,
    int N)
{
    __shared__ __align__(16) int   s_ch[2][BLK][STRIALS];
    __shared__ __align__(16) int   s_cu[2][BLK][STRIALS];
    __shared__ __align__(16) float s_rw[2][BLK][STRIALS];

    const int tid = threadIdx.x;
    const int n   = blockIdx.x * BLK + tid;
    if (n >= N) return;

    // scalar parameters (uniform; one-time transcendental setup lands on the
    // scalar float pipe)
    const float a_p  = 1.0f / (1.0f + __expf(-p_ap[0]));   // sigmoid(alpha_plus)
    const float a_m  = 1.0f / (1.0f + __expf(-p_am[0]));   // sigmoid(alpha_minus)
    const float b_i  = p_bi[0];
    const float b_s  = p_bs[0];
    const float b_t  = p_bt[0];
    const float vini = 100.0f * tanhf(p_iv[0]);            // MAX_INIT * tanh(iv)

    const char* gch = (const char*)(choice     + (size_t)n * TT);
    const char* gcu = (const char*)(choice_upd + (size_t)n * TT);
    const char* grw = (const char*)(reward     + (size_t)n * TT);
    v4f*        o   = (v4f*)       (out        + (size_t)n * TT * KK);

    // recurrent state
    float v0 = vini, v1 = vini, v2 = vini;   // RW values
    float c0 = 0.f,  c1 = 0.f,  c2 = 0.f;    // exact choice counts (f32)
    float sk0 = 0.f, sk1 = 0.f, sk2 = 0.f;   // stick add-in carried from t-1

    // ---- issue stage 0 into buffer 0
    {
        char* lch = (char*)&s_ch[0][tid][0];
        char* lcu = (char*)&s_cu[0][tid][0];
        char* lrw = (char*)&s_rw[0][tid][0];
        #pragma unroll
        for (int j = 0; j < SCHUNK; ++j) {
            ASYNC_COPY16(gch + 16*j, lch + 16*j);
            ASYNC_COPY16(gcu + 16*j, lcu + 16*j);
            ASYNC_COPY16(grw + 16*j, lrw + 16*j);
        }
    }

    for (int s = 0; s < NSTAGE; ++s) {
        // ---- issue stage s+1 into the other buffer (fire and forget), then
        //      drain the 30 oldest ops (stage s) before consuming it
        if (s + 1 < NSTAGE) {
            const int   nb  = (s + 1) & 1;
            const char* pch = gch + (size_t)(s + 1) * STRIALS * 4;
            const char* pcu = gcu + (size_t)(s + 1) * STRIALS * 4;
            const char* prw = grw + (size_t)(s + 1) * STRIALS * 4;
            char* lch = (char*)&s_ch[nb][tid][0];
            char* lcu = (char*)&s_cu[nb][tid][0];
            char* lrw = (char*)&s_rw[nb][tid][0];
            #pragma unroll
            for (int j = 0; j < SCHUNK; ++j) {
                ASYNC_COPY16(pch + 16*j, lch + 16*j);
                ASYNC_COPY16(pcu + 16*j, lcu + 16*j);
                ASYNC_COPY16(prw + 16*j, lrw + 16*j);
            }
            WAIT_ASYNC_LE(AOPS);
        } else {
            WAIT_ASYNC_LE(0);      // final stage: drain everything
        }

        const int buf = s & 1;
        const int*   lch = &s_ch[buf][tid][0];
        const int*   lcu = &s_cu[buf][tid][0];
        const float* lrw = &s_rw[buf][tid][0];

        #pragma unroll
        for (int ck = 0; ck < SCHUNK; ++ck) {
            const v4i C = *(const v4i*)(lch + 4 * ck);   // ds_load_b128
            const v4i U = *(const v4i*)(lcu + 4 * ck);
            const v4f R = *(const v4f*)(lrw + 4 * ck);

            const int   cc[4] = {C.x, C.y, C.z, C.w};
            const int   uu[4] = {U.x, U.y, U.z, U.w};
            const float rr[4] = {R.x, R.y, R.z, R.w};

            float ob[12];
            #pragma unroll
            for (int i = 0; i < 4; ++i) {
                // ---- emit: pre-update values + info(counts) + stick(carried)
                ob[3*i+0] = fmaf(v0, b_t, fmaf(c0, b_i, sk0));
                ob[3*i+1] = fmaf(v1, b_t, fmaf(c1, b_i, sk1));
                ob[3*i+2] = fmaf(v2, b_t, fmaf(c2, b_i, sk2));

                // ---- info/stick state update; one compare per k feeds both
                const int  c  = cc[i];
                const bool e0 = (c == 0), e1 = (c == 1), e2 = (c == 2);
                c0 += e0 ? 1.f : 0.f;  sk0 = e0 ? b_s : 0.f;
                c1 += e1 ? 1.f : 0.f;  sk1 = e1 ? b_s : 0.f;
                c2 += e2 ? 1.f : 0.f;  sk2 = e2 ? b_s : 0.f;

                // ---- RW value update
                const int   u = uu[i];
                const float r = rr[i];
                const float chosen = (u == 0) ? v0 : ((u == 1) ? v1 : v2);
                const float pe     = (r == r) ? (r - chosen) : 0.f;  // NaN -> 0
                const float d      = ((pe >= 0.f) ? a_p : a_m) * pe;
                v0 += (u == 0) ? d : 0.f;
                v1 += (u == 1) ? d : 0.f;
                v2 += (u == 2) ? d : 0.f;
            }

            // 12 floats = 3 aligned 128-bit nontemporal streaming stores
            v4f ob0 = { ob[0], ob[1], ob[2],  ob[3] };
            v4f ob1 = { ob[4], ob[5], ob[6],  ob[7] };
            v4f ob2 = { ob[8], ob[9], ob[10], ob[11] };
            const int cg = s * SCHUNK + ck;
            __builtin_nontemporal_store(ob0, &o[cg * 3 + 0]);
            __builtin_nontemporal_store(ob1, &o[cg * 3 + 1]);
            __builtin_nontemporal_store(ob2, &o[cg * 3 + 2]);
        }
    }
}

extern "C" void kernel_launch(void* const* d_in, const int* in_sizes, int n_in,
                              void* d_out, int out_size, void* d_ws, size_t ws_size,
                              hipStream_t stream) {
    (void)n_in; (void)out_size; (void)d_ws; (void)ws_size;

    const int*   choice     = (const int*)  d_in[0];
    const int*   choice_upd = (const int*)  d_in[1];
    const float* reward     = (const float*)d_in[2];
    const float* p_ap       = (const float*)d_in[3];
    const float* p_am       = (const float*)d_in[4];
    const float* p_iv       = (const float*)d_in[5];
    const float* p_bi       = (const float*)d_in[6];
    const float* p_bs       = (const float*)d_in[7];
    const float* p_bt       = (const float*)d_in[8];
    float*       out        = (float*)d_out;

    const int N = in_sizes[0] / TT;   // 8192

    const int grid = (N + BLK - 1) / BLK;   // 256 single-wave workgroups
    rw_scan_kernel<<<grid, BLK, 0, stream>>>(
        choice, choice_upd, reward,
        p_ap, p_am, p_iv, p_bi, p_bs, p_bt,
        out, N);
}